// MamboPooling_16209206575152
// MI455X (gfx1250) — compile-verified
//
#include <hip/hip_runtime.h>
#include <math.h>

#define N_NODES 100000
#define N_EDGES 3200000
#define DIM     128
#define K_KEEP  50000
#define NBLK    ((N_NODES + 255) / 256)   // 391

#define ROWS_PER_BLOCK 160               // 10 strips of 16; 100000 = 625 * 160
#define LDS_PITCH      130               // transposed W pitch: conflict-free, 8B aligned

typedef __attribute__((ext_vector_type(2))) float v2f;
typedef __attribute__((ext_vector_type(8))) float v8f;

// ---------------------------------------------------------------------------
// 1) h = x @ W via V_WMMA_F32_16X16X4_F32.
//    Block = 256 threads (8 waves). W staged TRANSPOSED + padded in LDS:
//    Wt[col*130 + k] -> each B fragment is one aligned ds_load_b64, banks at
//    stride 2 across lanes (conflict-free). Each wave hoists all 32 B
//    fragments into registers once, then runs 10 row-strips of 16 rows:
//    steady state is pure {global_load_b64 A, v_wmma} chains.
//    f32 A 16x4 layout: lanes 0-15 hold M=lane (K=k0,k0+1), lanes 16-31 hold
//    M=lane-16 (K=k0+2,k0+3). C/D: VGPR r -> M=r+8*half, N=lane&15.
// ---------------------------------------------------------------------------
__global__ __launch_bounds__(256) void gemm_wmma_f32(const float* __restrict__ x,
                                                     const float* __restrict__ W,
                                                     float* __restrict__ h) {
    extern __shared__ float Wl[];   // DIM * LDS_PITCH floats (transposed, padded)

    // cooperative transpose-load: coalesced global float4 reads, scalar LDS writes
    for (int i = threadIdx.x; i < DIM * DIM / 4; i += 256) {
        const int idx = i * 4;
        const int k   = idx / DIM;       // W row  (K index)
        const int col = idx % DIM;       // W col  (N index)
        float4 w = ((const float4*)W)[i];
        Wl[(col + 0) * LDS_PITCH + k] = w.x;
        Wl[(col + 1) * LDS_PITCH + k] = w.y;
        Wl[(col + 2) * LDS_PITCH + k] = w.z;
        Wl[(col + 3) * LDS_PITCH + k] = w.w;
    }
    __syncthreads();

    const int wave = threadIdx.x >> 5;
    const int lane = threadIdx.x & 31;
    const int half = lane >> 4;      // 0: K pair (k0,k0+1), 1: K pair (k0+2,k0+3)
    const int l16  = lane & 15;
    const int col0 = wave * 16;
    const int rowBase = blockIdx.x * ROWS_PER_BLOCK;

    // hoist all 32 B fragments (depend only on k and col) into registers
    v2f bf[DIM / 4];
#pragma unroll
    for (int kk = 0; kk < DIM / 4; ++kk) {
        const int ka = kk * 4 + 2 * half;
        bf[kk] = *(const v2f*)(&Wl[(col0 + l16) * LDS_PITCH + ka]);
    }

    for (int s = 0; s < ROWS_PER_BLOCK / 16; ++s) {
        const int row0 = rowBase + s * 16;
        const float* xrow = x + (size_t)(row0 + l16) * DIM;

        v8f acc = {};
#pragma unroll
        for (int kk = 0; kk < DIM / 4; ++kk) {
            const int ka = kk * 4 + 2 * half;
            v2f a = *(const v2f*)(xrow + ka);   // A[M=l16][ka], A[M=l16][ka+1]
            acc = __builtin_amdgcn_wmma_f32_16x16x4_f32(
                /*neg_a=*/false, a, /*neg_b=*/false, bf[kk],
                /*c_mod=*/(short)0, acc, /*reuse_a=*/false, /*reuse_b=*/false);
        }

        float* hout = h + (size_t)row0 * DIM + col0 + l16;
#pragma unroll
        for (int r = 0; r < 8; ++r) {
            const int m = r + 8 * half;
            hout[(size_t)m * DIM] = acc[r];
        }
    }
}

// ---------------------------------------------------------------------------
// 2) Degree / normalization / aggregation
// ---------------------------------------------------------------------------
__global__ __launch_bounds__(256) void zero_f32(float* __restrict__ p, int n) {
    int i = blockIdx.x * 256 + threadIdx.x;
    if (i < n) p[i] = 0.0f;
}

__global__ __launch_bounds__(256) void deg_accum(const int* __restrict__ dst,
                                                 float* __restrict__ deg) {
    int e = blockIdx.x * 256 + threadIdx.x;
    if (e < N_EDGES) atomicAdd(&deg[dst[e]], 1.0f);
}

__global__ __launch_bounds__(256) void make_dinv(const float* __restrict__ deg,
                                                 float* __restrict__ dinv) {
    int i = blockIdx.x * 256 + threadIdx.x;
    if (i < N_NODES) dinv[i] = rsqrtf(deg[i] + 1.0f);
}

// self-loop term: agg = h * dinv^2   (one lane-quad of 4 floats per thread)
__global__ __launch_bounds__(256) void agg_init(const float* __restrict__ h,
                                                const float* __restrict__ dinv,
                                                float* __restrict__ agg) {
    int gid = blockIdx.x * 256 + threadIdx.x;
    int n = gid >> 5, q = gid & 31;
    if (n >= N_NODES) return;
    float s = dinv[n]; s = s * s;
    float4 v = ((const float4*)(h + (size_t)n * DIM))[q];
    v.x *= s; v.y *= s; v.z *= s; v.w *= s;
    ((float4*)(agg + (size_t)n * DIM))[q] = v;
}

// one wave per edge: 32 lanes x float4 = the whole 128-float row (L2-resident)
__global__ __launch_bounds__(256) void edge_agg(const int* __restrict__ src,
                                                const int* __restrict__ dst,
                                                const float* __restrict__ h,
                                                const float* __restrict__ dinv,
                                                float* __restrict__ agg) {
    long long gid = (long long)blockIdx.x * 256 + threadIdx.x;
    int e = (int)(gid >> 5), q = (int)(gid & 31);
    if (e >= N_EDGES) return;
    int s = src[e], t = dst[e];
    float nrm = dinv[s] * dinv[t];
    float4 v = ((const float4*)(h + (size_t)s * DIM))[q];
    float* a = agg + (size_t)t * DIM + q * 4;
    atomicAdd(a + 0, v.x * nrm);
    atomicAdd(a + 1, v.y * nrm);
    atomicAdd(a + 2, v.z * nrm);
    atomicAdd(a + 3, v.w * nrm);
}

// agg += bias (in place -> final GCN output), score = ||row||_2 (wave reduce)
__global__ __launch_bounds__(256) void finalize_score(float* __restrict__ agg,
                                                      const float* __restrict__ bias,
                                                      float* __restrict__ score) {
    int gid = blockIdx.x * 256 + threadIdx.x;
    int n = gid >> 5, q = gid & 31;
    if (n >= N_NODES) return;
    float4 v = ((float4*)(agg + (size_t)n * DIM))[q];
    float4 b = ((const float4*)bias)[q];
    v.x += b.x; v.y += b.y; v.z += b.z; v.w += b.w;
    ((float4*)(agg + (size_t)n * DIM))[q] = v;
    float ss = v.x * v.x + v.y * v.y + v.z * v.z + v.w * v.w;
    for (int off = 16; off > 0; off >>= 1) ss += __shfl_xor(ss, off, 32);
    if (q == 0) score[n] = sqrtf(ss + 1e-12f);
}

// ---------------------------------------------------------------------------
// 3) Deterministic top-K: radix select threshold on float bits (scores >= 0
//    so the uint bit pattern is order-preserving), then scan-based compaction.
// ---------------------------------------------------------------------------
__global__ __launch_bounds__(1024) void radix_select(const float* __restrict__ score,
                                                     unsigned* __restrict__ sel) {
    __shared__ int hist[256];
    __shared__ unsigned s_prefix;
    __shared__ int s_want, s_above;
    if (threadIdx.x == 0) { s_prefix = 0u; s_want = K_KEEP; s_above = 0; }
    __syncthreads();

    for (int byte = 3; byte >= 0; --byte) {
        if (threadIdx.x < 256) hist[threadIdx.x] = 0;
        __syncthreads();
        const unsigned prefix = s_prefix;
        const unsigned umask  = (byte == 3) ? 0u : (0xFFFFFFFFu << (8 * (byte + 1)));
        for (int i = threadIdx.x; i < N_NODES; i += 1024) {
            unsigned bits = __float_as_uint(score[i]);
            if ((bits & umask) == (prefix & umask))
                atomicAdd(&hist[(bits >> (8 * byte)) & 0xFF], 1);
        }
        __syncthreads();
        if (threadIdx.x == 0) {
            int want = s_want, cum = 0, bin;
            for (bin = 255; bin > 0; --bin) {
                if (cum + hist[bin] >= want) break;
                cum += hist[bin];
            }
            s_above  = s_above + cum;           // strictly greater so far
            s_want   = want - cum;              // still needed inside chosen bin
            s_prefix = prefix | ((unsigned)bin << (8 * byte));
        }
        __syncthreads();
    }
    if (threadIdx.x == 0) { sel[0] = s_prefix; sel[1] = (unsigned)s_above; }
}

__global__ __launch_bounds__(256) void flag_block_sums(const float* __restrict__ score,
                                                       const unsigned* __restrict__ sel,
                                                       int* __restrict__ bsumG,
                                                       int* __restrict__ bsumE) {
    __shared__ int cg, ce;
    if (threadIdx.x == 0) { cg = 0; ce = 0; }
    __syncthreads();
    int n = blockIdx.x * 256 + threadIdx.x;
    if (n < N_NODES) {
        unsigned T = sel[0];
        unsigned b = __float_as_uint(score[n]);
        if (b > T) atomicAdd(&cg, 1);
        else if (b == T) atomicAdd(&ce, 1);
    }
    __syncthreads();
    if (threadIdx.x == 0) { bsumG[blockIdx.x] = cg; bsumE[blockIdx.x] = ce; }
}

__global__ __launch_bounds__(512) void scan_block_sums(const int* __restrict__ bsumG,
                                                       const int* __restrict__ bsumE,
                                                       int* __restrict__ boffG,
                                                       int* __restrict__ boffE) {
    __shared__ int sg[512], se[512];
    int t = threadIdx.x;
    sg[t] = (t < NBLK) ? bsumG[t] : 0;
    se[t] = (t < NBLK) ? bsumE[t] : 0;
    __syncthreads();
    for (int off = 1; off < 512; off <<= 1) {
        int vg = (t >= off) ? sg[t - off] : 0;
        int ve = (t >= off) ? se[t - off] : 0;
        __syncthreads();
        sg[t] += vg; se[t] += ve;
        __syncthreads();
    }
    if (t < NBLK) { boffG[t] = sg[t] - bsumG[t]; boffE[t] = se[t] - bsumE[t]; }
}

__global__ __launch_bounds__(256) void assign_pos(const float* __restrict__ score,
                                                  const unsigned* __restrict__ sel,
                                                  const int* __restrict__ boffG,
                                                  const int* __restrict__ boffE,
                                                  int* __restrict__ pos) {
    __shared__ int sg[256], se[256];
    int t = threadIdx.x;
    int n = blockIdx.x * 256 + t;
    unsigned T = sel[0];
    int countG = (int)sel[1];
    int g = 0, e = 0;
    if (n < N_NODES) {
        unsigned b = __float_as_uint(score[n]);
        g = (b > T) ? 1 : 0;
        e = (b == T) ? 1 : 0;
    }
    sg[t] = g; se[t] = e;
    __syncthreads();
    for (int off = 1; off < 256; off <<= 1) {
        int vg = (t >= off) ? sg[t - off] : 0;
        int ve = (t >= off) ? se[t - off] : 0;
        __syncthreads();
        sg[t] += vg; se[t] += ve;
        __syncthreads();
    }
    if (n < N_NODES) {
        int p = -1;
        int exG = sg[t] - g, exE = se[t] - e;
        if (g) p = boffG[blockIdx.x] + exG;
        else if (e) {
            int re = boffE[blockIdx.x] + exE;
            if (re < K_KEEP - countG) p = countG + re;
        }
        pos[n] = p;
    }
}

__global__ __launch_bounds__(256) void gather_out(const float* __restrict__ hg,
                                                  const float* __restrict__ score,
                                                  const int* __restrict__ pos,
                                                  float* __restrict__ xout) {
    int gid = blockIdx.x * 256 + threadIdx.x;
    int n = gid >> 5, q = gid & 31;
    if (n >= N_NODES) return;
    int p = pos[n];
    if (p < 0) return;
    float gate = tanhf(score[n]);
    float4 v = ((const float4*)(hg + (size_t)n * DIM))[q];
    v.x *= gate; v.y *= gate; v.z *= gate; v.w *= gate;
    ((float4*)(xout + (size_t)p * DIM))[q] = v;
}

__global__ __launch_bounds__(256) void remap_edges(const int* __restrict__ src,
                                                   const int* __restrict__ dst,
                                                   const int* __restrict__ pos,
                                                   int* __restrict__ eout,
                                                   float* __restrict__ mout) {
    int e = blockIdx.x * 256 + threadIdx.x;
    if (e >= N_EDGES) return;
    int ns = pos[src[e]], nt = pos[dst[e]];
    bool m = (ns >= 0) && (nt >= 0);
    eout[e]           = m ? ns : -1;
    eout[N_EDGES + e] = m ? nt : -1;
    mout[e]           = m ? 1.0f : 0.0f;
}

// ---------------------------------------------------------------------------
extern "C" void kernel_launch(void* const* d_in, const int* in_sizes, int n_in,
                              void* d_out, int out_size, void* d_ws, size_t ws_size,
                              hipStream_t stream) {
    const float* x    = (const float*)d_in[0];
    const int*   eidx = (const int*)d_in[1];
    const float* W    = (const float*)d_in[2];
    const float* bias = (const float*)d_in[3];
    const int* src = eidx;
    const int* dst = eidx + N_EDGES;

    // workspace partition
    float* h     = (float*)d_ws;                       // N*DIM
    float* agg   = h + (size_t)N_NODES * DIM;          // N*DIM  (becomes final GCN out)
    float* deg   = agg + (size_t)N_NODES * DIM;        // N
    float* dinv  = deg + N_NODES;                      // N
    float* score = dinv + N_NODES;                     // N
    int*   pos   = (int*)(score + N_NODES);            // N
    unsigned* sel = (unsigned*)(pos + N_NODES);        // 8
    int* bsumG = (int*)(sel + 8);                      // NBLK each
    int* bsumE = bsumG + NBLK;
    int* boffG = bsumE + NBLK;
    int* boffE = boffG + NBLK;

    // output partition: [x_out (K*DIM f32) | new_edge_index (2E i32) | mask (E)]
    float* outF = (float*)d_out;
    float* xout = outF;
    int*   eout = (int*)(outF + (size_t)K_KEEP * DIM);
    float* mout = outF + (size_t)K_KEEP * DIM + 2 * (size_t)N_EDGES;

    // 1) GEMM: h = x @ W  (WMMA f32 16x16x4, transposed W in LDS, B in regs)
    gemm_wmma_f32<<<N_NODES / ROWS_PER_BLOCK, 256,
                    DIM * LDS_PITCH * sizeof(float), stream>>>(x, W, h);

    // 2) degrees + normalization
    zero_f32<<<(N_NODES + 255) / 256, 256, 0, stream>>>(deg, N_NODES);
    deg_accum<<<(N_EDGES + 255) / 256, 256, 0, stream>>>(dst, deg);
    make_dinv<<<(N_NODES + 255) / 256, 256, 0, stream>>>(deg, dinv);

    // 3) aggregation (self-loop init, then edge scatter-add; all L2-resident)
    agg_init<<<(N_NODES * 32 + 255) / 256, 256, 0, stream>>>(h, dinv, agg);
    edge_agg<<<(int)(((long long)N_EDGES * 32 + 255) / 256), 256, 0, stream>>>(src, dst, h, dinv, agg);
    finalize_score<<<(N_NODES * 32 + 255) / 256, 256, 0, stream>>>(agg, bias, score);

    // 4) deterministic top-K selection + compaction
    radix_select<<<1, 1024, 0, stream>>>(score, sel);
    flag_block_sums<<<NBLK, 256, 0, stream>>>(score, sel, bsumG, bsumE);
    scan_block_sums<<<1, 512, 0, stream>>>(bsumG, bsumE, boffG, boffE);
    assign_pos<<<NBLK, 256, 0, stream>>>(score, sel, boffG, boffE, pos);

    // 5) gated gather of kept nodes + edge remap
    gather_out<<<(N_NODES * 32 + 255) / 256, 256, 0, stream>>>(agg, score, pos, xout);
    remap_edges<<<(N_EDGES + 255) / 256, 256, 0, stream>>>(src, dst, pos, eout, mout);

    (void)in_sizes; (void)n_in; (void)out_size; (void)ws_size;
}